// HTMC_83296595739033
// MI455X (gfx1250) — compile-verified
//
#include <hip/hip_runtime.h>
#include <math.h>

// ---------------------------------------------------------------------------
// DGCNN-style point matcher for MI455X (gfx1250, wave32).
// All heavy 1x1-conv layers run on V_WMMA_F32_16X16X4_F32 (fp32 WMMA) so the
// fp32 reference numerics are preserved. The 419MB fusion tensor is never
// materialized: decode builds the B-operand as s[c,ns]*t[c,nt] on the fly,
// with 16 accumulator tiles register-blocked so each A/s load feeds 16 WMMAs.
// ---------------------------------------------------------------------------

#define NPTS 320
#define KNN  20
#define KPAD 32   // k padded to 2 WMMA column tiles

typedef __attribute__((ext_vector_type(2))) float v2f;
typedef __attribute__((ext_vector_type(8))) float v8f;

__device__ __forceinline__ v8f zero8() {
  v8f z;
#pragma unroll
  for (int i = 0; i < 8; ++i) z[i] = 0.f;
  return z;
}

// D = A(16x4) * B(4x16) + C, fp32 WMMA.  8-arg VOP3P form:
// (neg_a, A, neg_b, B, c_mod, C, reuse_a, reuse_b)
__device__ __forceinline__ v8f wmma4(v2f a, v2f b, v8f c) {
  return __builtin_amdgcn_wmma_f32_16x16x4_f32(false, a, false, b, (short)0, c,
                                               false, false);
}

// A-matrix 16x4 tile from row-major W[ld] at (o0,k0).
// ISA layout: lanes 0-15 hold M=0..15 with K={k0,k0+1}; lanes 16-31 K={k0+2,k0+3}.
__device__ __forceinline__ v2f load_a(const float* __restrict__ W, int ld,
                                      int o0, int k0, int lane) {
  int m  = lane & 15;
  int kk = k0 + ((lane >> 4) << 1);
  v2f a;
  a.x = W[(o0 + m) * ld + kk];
  a.y = W[(o0 + m) * ld + kk + 1];
  return a;
}

// B-matrix 4x16 tile from LDS X (row stride S) at (k0,n0); lanes mirror A.
__device__ __forceinline__ v2f load_b(const float* __restrict__ X, int S,
                                      int k0, int n0, int lane) {
  int n  = lane & 15;
  int kk = k0 + ((lane >> 4) << 1);
  v2f b;
  b.x = X[kk * S + n0 + n];
  b.y = X[(kk + 1) * S + n0 + n];
  return b;
}

// Store C/D tile (VGPR v -> rows o0+v and o0+v+8) with BN(g,b)+ReLU into LDS.
__device__ __forceinline__ void store_c_bnrelu(float* __restrict__ H, int S,
                                               int o0, int n0,
                                               const float* __restrict__ g,
                                               const float* __restrict__ bb,
                                               v8f c, int lane) {
  int n  = lane & 15;
  int mo = (lane >> 4) << 3;  // 0 or 8
#pragma unroll
  for (int v = 0; v < 8; ++v) {
    int ch = o0 + v + mo;
    float val = g[ch] * c[v] + bb[ch];
    H[ch * S + n0 + n] = fmaxf(val, 0.f);
  }
}

// ---------------------------------------------------------------------------
// kNN: one wave per (b,n).  neg_dist = -||x_n - x_m||^2, pick top-20 indices.
// ---------------------------------------------------------------------------
__global__ __launch_bounds__(32) void knn_kernel(const float* __restrict__ x,
                                                 int* __restrict__ idx_out) {
  __shared__ float d[NPTS];
  __shared__ float bv[32];
  __shared__ int bi[32];
  int b = blockIdx.x / NPTS, n = blockIdx.x % NPTS;
  int tid = threadIdx.x;
  float x0 = x[(b * 3 + 0) * NPTS + n];
  float x1 = x[(b * 3 + 1) * NPTS + n];
  float x2 = x[(b * 3 + 2) * NPTS + n];
  for (int m = tid; m < NPTS; m += 32) {
    float d0 = x0 - x[(b * 3 + 0) * NPTS + m];
    float d1 = x1 - x[(b * 3 + 1) * NPTS + m];
    float d2 = x2 - x[(b * 3 + 2) * NPTS + m];
    d[m] = -(d0 * d0 + d1 * d1 + d2 * d2);
  }
  __syncthreads();
  for (int s = 0; s < KNN; ++s) {
    float best = -3.4e38f;
    int besti = 0;
    for (int m = tid; m < NPTS; m += 32)
      if (d[m] > best) { best = d[m]; besti = m; }
    bv[tid] = best;
    bi[tid] = besti;
    __syncthreads();
    if (tid == 0) {
      float bbv = bv[0];
      int ii = bi[0];
      for (int t = 1; t < 32; ++t)
        if (bv[t] > bbv || (bv[t] == bbv && bi[t] < ii)) { bbv = bv[t]; ii = bi[t]; }
      idx_out[(b * NPTS + n) * KNN + s] = ii;
      d[ii] = -3.4e38f;
    }
    __syncthreads();
  }
}

// ---------------------------------------------------------------------------
// Fused encoder: per point n, gather 20 neighbors -> feat[6,20] -> 64 -> 256
// -> 512 -> max over k.  Layers 1-3 on fp32 WMMA; layer-3 accumulators stay
// in registers; max-over-k via half-wave shfl reductions.
// ---------------------------------------------------------------------------
__global__ __launch_bounds__(256) void encode_kernel(
    const float* __restrict__ pts, const int* __restrict__ idx,
    const float* __restrict__ W1, const float* __restrict__ g1, const float* __restrict__ b1,
    const float* __restrict__ W2, const float* __restrict__ g2, const float* __restrict__ b2,
    const float* __restrict__ W3, const float* __restrict__ g3, const float* __restrict__ b3,
    float* __restrict__ feat_out) {
  __shared__ float w1p[64 * 8];       // 2KB  zero-padded W1 (K 6->8)
  __shared__ float feat[8 * KPAD];    // 1KB  (K padded 6->8, k padded 20->32)
  __shared__ float h1[64 * KPAD];     // 8KB
  __shared__ float h2[256 * KPAD];    // 32KB
  const int b = blockIdx.x / NPTS;
  const int n = blockIdx.x % NPTS;
  const int tid = threadIdx.x;
  const int lane = tid & 31;
  const int wave = tid >> 5;

  for (int t = tid; t < 8 * KPAD; t += 256) feat[t] = 0.f;
  for (int t = tid; t < 64 * 8; t += 256) {
    int o = t >> 3, k = t & 7;
    w1p[t] = (k < 6) ? W1[o * 6 + k] : 0.f;
  }
  __syncthreads();
  if (tid < KNN) {
    int nbr = idx[(b * NPTS + n) * KNN + tid];
#pragma unroll
    for (int c = 0; c < 3; ++c) {
      feat[c * KPAD + tid]       = pts[(b * 3 + c) * NPTS + nbr];  // neighbor
      feat[(3 + c) * KPAD + tid] = pts[(b * 3 + c) * NPTS + n];    // center
    }
  }
  __syncthreads();

  // Layer 1: h1[64,32] = relu(bn(W1[64,6] @ feat))   (K padded to 8)
  {
    int r = wave >> 1, ct = wave & 1;
    int o0 = r * 16, n0 = ct * 16;
    v8f acc = zero8();
#pragma unroll
    for (int k0 = 0; k0 < 8; k0 += 4) {
      v2f a = load_a(w1p, 8, o0, k0, lane);       // LDS, no exec branches
      v2f bb = load_b(feat, KPAD, k0, n0, lane);
      acc = wmma4(a, bb, acc);
    }
    store_c_bnrelu(h1, KPAD, o0, n0, g1, b1, acc, lane);
  }
  __syncthreads();

  // Layer 2: h2[256,32] = relu(bn(W2[256,64] @ h1))
#pragma unroll
  for (int i = 0; i < 4; ++i) {
    int t = wave + i * 8;
    int o0 = (t >> 1) * 16, n0 = (t & 1) * 16;
    v8f acc = zero8();
    for (int k0 = 0; k0 < 64; k0 += 4) {
      v2f a = load_a(W2, 64, o0, k0, lane);
      v2f bb = load_b(h1, KPAD, k0, n0, lane);
      acc = wmma4(a, bb, acc);
    }
    store_c_bnrelu(h2, KPAD, o0, n0, g2, b2, acc, lane);
  }
  __syncthreads();

  // Layer 3 + max over k (in registers, no 64KB h3 buffer needed)
  for (int i = 0; i < 4; ++i) {
    int o0 = (wave + i * 8) * 16;
    __builtin_prefetch(&W3[o0 * 256], 0, 3);
    v8f a0 = zero8(), a1 = zero8();
    for (int k0 = 0; k0 < 256; k0 += 4) {
      v2f a = load_a(W3, 256, o0, k0, lane);
      v2f bc0 = load_b(h2, KPAD, k0, 0, lane);
      v2f bc1 = load_b(h2, KPAD, k0, 16, lane);
      a0 = wmma4(a, bc0, a0);
      a1 = wmma4(a, bc1, a1);
    }
    int kn = lane & 15;          // k within column tile
    int mo = (lane >> 4) << 3;   // row offset 0/8
#pragma unroll
    for (int v = 0; v < 8; ++v) {
      int ch = o0 + v + mo;
      float gg = g3[ch], bb2 = b3[ch];
      float m0 = fmaxf(gg * a0[v] + bb2, 0.f);                          // k = kn
      float m1 = (kn < KNN - 16) ? fmaxf(gg * a1[v] + bb2, 0.f) : 0.f;  // k = 16+kn
      float mx = fmaxf(m0, m1);
#pragma unroll
      for (int off = 1; off < 16; off <<= 1)
        mx = fmaxf(mx, __shfl_xor(mx, off, 32));
      if (kn == 0) feat_out[(b * 512 + ch) * NPTS + n] = mx;
    }
  }
}

// ---------------------------------------------------------------------------
// Global head: 512 -> 256 -> 128 -> 64, max over N.  Tiny (0.1 GFLOP), plain
// VALU with LDS chunking.
// ---------------------------------------------------------------------------
__global__ __launch_bounds__(256) void global_kernel(
    const float* __restrict__ feat,
    const float* __restrict__ W1, const float* __restrict__ g1, const float* __restrict__ b1,
    const float* __restrict__ W2, const float* __restrict__ g2, const float* __restrict__ b2,
    const float* __restrict__ W3, const float* __restrict__ g3, const float* __restrict__ b3,
    float* __restrict__ gout) {
  __shared__ float h1[256 * 32];  // 32KB
  __shared__ float h2[128 * 32];  // 16KB
  __shared__ float h3[64 * 32];   // 8KB
  __shared__ float gmax[64];
  int b = blockIdx.x, tid = threadIdx.x;
  if (tid < 64) gmax[tid] = -3.4e38f;
  __syncthreads();
  for (int n0 = 0; n0 < NPTS; n0 += 32) {
    for (int t = tid; t < 256 * 32; t += 256) {
      int o = t >> 5, j = t & 31;
      float acc = 0.f;
      for (int c = 0; c < 512; ++c)
        acc += W1[o * 512 + c] * feat[(b * 512 + c) * NPTS + n0 + j];
      h1[t] = fmaxf(g1[o] * acc + b1[o], 0.f);
    }
    __syncthreads();
    for (int t = tid; t < 128 * 32; t += 256) {
      int o = t >> 5, j = t & 31;
      float acc = 0.f;
      for (int c = 0; c < 256; ++c) acc += W2[o * 256 + c] * h1[c * 32 + j];
      h2[t] = fmaxf(g2[o] * acc + b2[o], 0.f);
    }
    __syncthreads();
    for (int t = tid; t < 64 * 32; t += 256) {
      int o = t >> 5, j = t & 31;
      float acc = 0.f;
      for (int c = 0; c < 128; ++c) acc += W3[o * 128 + c] * h2[c * 32 + j];
      h3[t] = fmaxf(g3[o] * acc + b3[o], 0.f);
    }
    __syncthreads();
    if (tid < 64) {
      float m = gmax[tid];
      for (int j = 0; j < 32; ++j) m = fmaxf(m, h3[tid * 32 + j]);
      gmax[tid] = m;
    }
    __syncthreads();
  }
  if (tid < 64) gout[b * 64 + tid] = gmax[tid];
}

// ---------------------------------------------------------------------------
// Mask head: gin[640,N]=concat(feat, g_self, g_other) -> 512 -> 128 -> 2 ->
// softmax[:,1].  WMMA, gin B-operand gathered on the fly (L2-resident).
// ---------------------------------------------------------------------------
__device__ __forceinline__ float gin_val(const float* __restrict__ feat,
                                         const float* __restrict__ gs,
                                         const float* __restrict__ go, int b,
                                         int c, int n) {
  if (c < 512) return feat[(b * 512 + c) * NPTS + n];
  if (c < 576) return gs[b * 64 + (c - 512)];
  return go[b * 64 + (c - 576)];
}

__global__ __launch_bounds__(256) void mask_kernel(
    const float* __restrict__ feat, const float* __restrict__ gs, const float* __restrict__ go,
    const float* __restrict__ W1, const float* __restrict__ g1, const float* __restrict__ b1,
    const float* __restrict__ W2, const float* __restrict__ g2, const float* __restrict__ b2,
    const float* __restrict__ W3, float* __restrict__ mask) {
  __shared__ float h1[512 * 16];  // 32KB
  __shared__ float h2[128 * 16];  // 8KB
  int b = blockIdx.x / (NPTS / 16);
  int n0 = (blockIdx.x % (NPTS / 16)) * 16;
  int tid = threadIdx.x, lane = tid & 31, wave = tid >> 5;

  for (int i = 0; i < 4; ++i) {  // Layer 1: 32 row tiles / 8 waves
    int o0 = (wave + i * 8) * 16;
    __builtin_prefetch(&W1[o0 * 640], 0, 3);
    v8f acc = zero8();
    int nn = lane & 15;
    for (int k0 = 0; k0 < 640; k0 += 4) {
      v2f a = load_a(W1, 640, o0, k0, lane);
      int kk = k0 + ((lane >> 4) << 1);
      v2f bb;
      bb.x = gin_val(feat, gs, go, b, kk, n0 + nn);
      bb.y = gin_val(feat, gs, go, b, kk + 1, n0 + nn);
      acc = wmma4(a, bb, acc);
    }
    store_c_bnrelu(h1, 16, o0, 0, g1, b1, acc, lane);
  }
  __syncthreads();
  {  // Layer 2
    int o0 = wave * 16;
    v8f acc = zero8();
    for (int k0 = 0; k0 < 512; k0 += 4) {
      v2f a = load_a(W2, 512, o0, k0, lane);
      v2f bb = load_b(h1, 16, k0, 0, lane);
      acc = wmma4(a, bb, acc);
    }
    store_c_bnrelu(h2, 16, o0, 0, g2, b2, acc, lane);
  }
  __syncthreads();
  if (tid < 16) {  // logits + softmax[:,1] = sigmoid(l1 - l0)
    float l0 = 0.f, l1 = 0.f;
    for (int c = 0; c < 128; ++c) {
      float h = h2[c * 16 + tid];
      l0 += W3[c] * h;
      l1 += W3[128 + c] * h;
    }
    mask[b * NPTS + n0 + tid] = 1.f / (1.f + expf(l0 - l1));
  }
}

// ---------------------------------------------------------------------------
// Decode (dominant ~30 GFLOP): per 16x16 (ns,nt) tile.  Register-blocked:
// 16 accumulator tiles (one per nt) so each A-tile / s-pair load feeds 16
// WMMAs; only the t products stream from LDS.  1 wmma : ~0.13 glb + 1 ds.
// ---------------------------------------------------------------------------
__global__ __launch_bounds__(256, 1) void decode_kernel(
    const float* __restrict__ sfeat, const float* __restrict__ tfeat,
    const float* __restrict__ W5, const float* __restrict__ g5, const float* __restrict__ b5,
    const float* __restrict__ W3, const float* __restrict__ g3, const float* __restrict__ b3,
    const float* __restrict__ W1, float* __restrict__ out) {
  __shared__ float t_lds[512 * 16];  // 32KB  t-tile [c][jt]
  __shared__ float h1[128 * 16];     // 8KB
  __shared__ float h2[64 * 16];      // 4KB
  const int TP = NPTS / 16;          // 20
  int b = blockIdx.x / (TP * TP);
  int rt = blockIdx.x % (TP * TP);
  int ns0 = (rt / TP) * 16;
  int nt0 = (rt % TP) * 16;
  int tid = threadIdx.x, lane = tid & 31, wave = tid >> 5;

  for (int t = tid; t < 512 * 16; t += 256) {
    int c = t >> 4, j = t & 15;
    t_lds[t] = tfeat[(b * 512 + c) * NPTS + nt0 + j];
  }
  __builtin_prefetch(W5, 0, 3);
  __syncthreads();

  // Layer 1: h1[128, 16ns, 16nt]; each wave owns one 16-row o-tile and all
  // 16 nt accumulators.  A (W5) and the s-pair are loaded once per k-step.
  const int o0 = wave * 16;
  const int nn = lane & 15;
  v8f acc[16];
#pragma unroll
  for (int j = 0; j < 16; ++j) acc[j] = zero8();
  for (int k0 = 0; k0 < 512; k0 += 4) {
    v2f a = load_a(W5, 512, o0, k0, lane);
    int kk = k0 + ((lane >> 4) << 1);
    v2f s;
    s.x = sfeat[(b * 512 + kk) * NPTS + ns0 + nn];
    s.y = sfeat[(b * 512 + kk + 1) * NPTS + ns0 + nn];
#pragma unroll
    for (int j = 0; j < 16; ++j) {
      v2f bb;
      bb.x = s.x * t_lds[kk * 16 + j];
      bb.y = s.y * t_lds[(kk + 1) * 16 + j];
      acc[j] = wmma4(a, bb, acc[j]);
    }
  }

  // Per-nt finalize: bn+relu -> h1, layer 2 (WMMA), layer 3 + softmax.
#pragma unroll
  for (int jt = 0; jt < 16; ++jt) {
    __syncthreads();  // previous iteration's layer-3 reads done
    store_c_bnrelu(h1, 16, o0, 0, g5, b5, acc[jt], lane);
    __syncthreads();
    if (wave < 4) {  // Layer 2: h2[64,16]
      int oo = wave * 16;
      v8f a2 = zero8();
      for (int k0 = 0; k0 < 128; k0 += 4) {
        v2f a = load_a(W3, 128, oo, k0, lane);
        v2f bb = load_b(h1, 16, k0, 0, lane);
        a2 = wmma4(a, bb, a2);
      }
      store_c_bnrelu(h2, 16, oo, 0, g3, b3, a2, lane);
    }
    __syncthreads();
    if (tid < 16) {
      float l0 = 0.f, l1 = 0.f;
      for (int c = 0; c < 64; ++c) {
        float h = h2[c * 16 + tid];
        l0 += W1[c] * h;
        l1 += W1[64 + c] * h;
      }
      out[(b * NPTS + ns0 + tid) * NPTS + nt0 + jt] = 1.f / (1.f + expf(l0 - l1));
    }
  }
}

// ---------------------------------------------------------------------------
// Host launcher.  Param pointers assume dict-insertion-order flattening of
// setup_inputs(): src, tgt, then We1,ge1,be1, We3,ge3,be3, We5,ge5,be5,
// Wg1,gg1,bg1, Wg2,gg2,bg2, Wg3,gg3,bg3, Wd5,gd5,bd5, Wd3,gd3,bd3, Wd1,
// Wm1,gm1,bm1, Wm2,gm2,bm2, Wm3.
// ---------------------------------------------------------------------------
extern "C" void kernel_launch(void* const* d_in, const int* in_sizes, int n_in,
                              void* d_out, int out_size, void* d_ws,
                              size_t ws_size, hipStream_t stream) {
  const int B = 2;
  const float* src = (const float*)d_in[0];
  const float* tgt = (const float*)d_in[1];
  const float* We1 = (const float*)d_in[2];
  const float* ge1 = (const float*)d_in[3];
  const float* be1 = (const float*)d_in[4];
  const float* We3 = (const float*)d_in[5];
  const float* ge3 = (const float*)d_in[6];
  const float* be3 = (const float*)d_in[7];
  const float* We5 = (const float*)d_in[8];
  const float* ge5 = (const float*)d_in[9];
  const float* be5 = (const float*)d_in[10];
  const float* Wg1 = (const float*)d_in[11];
  const float* gg1 = (const float*)d_in[12];
  const float* bg1 = (const float*)d_in[13];
  const float* Wg2 = (const float*)d_in[14];
  const float* gg2 = (const float*)d_in[15];
  const float* bg2 = (const float*)d_in[16];
  const float* Wg3 = (const float*)d_in[17];
  const float* gg3 = (const float*)d_in[18];
  const float* bg3 = (const float*)d_in[19];
  const float* Wd5 = (const float*)d_in[20];
  const float* gd5 = (const float*)d_in[21];
  const float* bd5 = (const float*)d_in[22];
  const float* Wd3 = (const float*)d_in[23];
  const float* gd3 = (const float*)d_in[24];
  const float* bd3 = (const float*)d_in[25];
  const float* Wd1 = (const float*)d_in[26];
  const float* Wm1 = (const float*)d_in[27];
  const float* gm1 = (const float*)d_in[28];
  const float* bm1 = (const float*)d_in[29];
  const float* Wm2 = (const float*)d_in[30];
  const float* gm2 = (const float*)d_in[31];
  const float* bm2 = (const float*)d_in[32];
  const float* Wm3 = (const float*)d_in[33];

  // workspace layout
  char* ws = (char*)d_ws;
  int* idx_s = (int*)ws;
  int* idx_t = idx_s + B * NPTS * KNN;
  float* sfeat = (float*)(idx_t + B * NPTS * KNN);
  float* tfeat = sfeat + B * 512 * NPTS;
  float* sglob = tfeat + B * 512 * NPTS;
  float* tglob = sglob + B * 64;

  float* corr = (float*)d_out;             // [B,N,N]
  float* smask = corr + B * NPTS * NPTS;   // [B,N]
  float* tmask = smask + B * NPTS;         // [B,N]

  knn_kernel<<<B * NPTS, 32, 0, stream>>>(src, idx_s);
  knn_kernel<<<B * NPTS, 32, 0, stream>>>(tgt, idx_t);

  encode_kernel<<<B * NPTS, 256, 0, stream>>>(src, idx_s, We1, ge1, be1, We3,
                                              ge3, be3, We5, ge5, be5, sfeat);
  encode_kernel<<<B * NPTS, 256, 0, stream>>>(tgt, idx_t, We1, ge1, be1, We3,
                                              ge3, be3, We5, ge5, be5, tfeat);

  global_kernel<<<B, 256, 0, stream>>>(sfeat, Wg1, gg1, bg1, Wg2, gg2, bg2,
                                       Wg3, gg3, bg3, sglob);
  global_kernel<<<B, 256, 0, stream>>>(tfeat, Wg1, gg1, bg1, Wg2, gg2, bg2,
                                       Wg3, gg3, bg3, tglob);

  mask_kernel<<<B * (NPTS / 16), 256, 0, stream>>>(
      sfeat, sglob, tglob, Wm1, gm1, bm1, Wm2, gm2, bm2, Wm3, smask);
  mask_kernel<<<B * (NPTS / 16), 256, 0, stream>>>(
      tfeat, tglob, sglob, Wm1, gm1, bm1, Wm2, gm2, bm2, Wm3, tmask);

  decode_kernel<<<B * (NPTS / 16) * (NPTS / 16), 256, 0, stream>>>(
      sfeat, tfeat, Wd5, gd5, bd5, Wd3, gd3, bd3, Wd1, corr);
}